// EmbeddingsAverage_13511967113310
// MI455X (gfx1250) — compile-verified
//
#include <hip/hip_runtime.h>
#include <hip/hip_bf16.h>

// Problem constants from the reference
#define BB   64
#define TT   2048
#define DD   1024
#define OUTN 8

typedef __attribute__((ext_vector_type(2))) float v2f;
typedef __attribute__((ext_vector_type(4))) float v4f;
typedef __attribute__((ext_vector_type(8))) float v8f;

// ---------------------------------------------------------------------------
// Kernel 1: ragged mean over T.  Bandwidth-bound: 512 MB of x, streamed once
// with non-temporal b128 loads.  Grid = (D/128, B); block = 256 threads:
//   lane (tid&31)  -> one float4 (covers 128 contiguous floats of d)
//   slice (tid>>5) -> 8 time-slices striding over T
// LDS reduce across the 8 slices (slice 0 sums them serially), then avg = sum / len.
// ---------------------------------------------------------------------------
__global__ void ragged_mean_kernel(const float* __restrict__ x,
                                   const int*   __restrict__ lengths,
                                   float*       __restrict__ avg) {
    const int b      = blockIdx.y;
    const int dbase  = blockIdx.x * 128;
    const int dlane  = threadIdx.x & 31;
    const int tslice = threadIdx.x >> 5;
    const int len    = lengths[b];

    const float* xb = x + (size_t)b * TT * DD + dbase + dlane * 4;

    v4f acc = {0.0f, 0.0f, 0.0f, 0.0f};
    #pragma unroll 4
    for (int t = tslice; t < len; t += 8) {
        // single-use stream: NT hint keeps the 512 MB of x out of the caches
        v4f v = __builtin_nontemporal_load((const v4f*)(xb + (size_t)t * DD));
        acc += v;
    }

    __shared__ v4f red[256];
    red[threadIdx.x] = acc;
    __syncthreads();

    if (tslice == 0) {
        v4f s = red[dlane];
        #pragma unroll
        for (int j = 1; j < 8; ++j) s += red[j * 32 + dlane];
        const float inv = 1.0f / (float)len;
        s *= inv;
        *(v4f*)(avg + (size_t)b * DD + dbase + dlane * 4) = s;
    }
}

// ---------------------------------------------------------------------------
// Kernel 2: out[64,8] = avg[64,1024] @ W.T[1024,8] + bias, via
// V_WMMA_F32_16X16X4_F32 (f32 in / f32 out, full precision).
// One workgroup, 4 waves; wave w owns M rows [16w, 16w+16).  N padded 8->16
// with zeros.  K loop: 1024/4 = 256 WMMA per wave.  Bias folded into C init.
//
// Fragment layouts (ISA 7.12.2):
//   A 16x4 f32, 2 VGPRs: lane l<16 -> A[l][0..1];  lane l>=16 -> A[l-16][2..3]
//   B 4x16,     2 VGPRs: v0 = rows K=0 (lanes 0-15) / K=2 (lanes 16-31); v1 = K=1/K=3
//   C/D 16x16,  8 VGPRs: VGPR v = row M=v (lanes 0-15, N=lane) / M=v+8 (lanes 16-31)
// ---------------------------------------------------------------------------
__global__ void gemm_wmma_f32_kernel(const float* __restrict__ avg,
                                     const float* __restrict__ W,
                                     const float* __restrict__ bias,
                                     float*       __restrict__ out) {
    __shared__ float wt[DD * OUTN];              // 32 KB: wt[k*8+n] = W[n][k]

    const int tid = threadIdx.x;                 // 128 threads = 4 waves
    for (int i = tid; i < DD * OUTN; i += 128) {
        const int k = i >> 3, n = i & 7;
        wt[i] = W[n * DD + k];
    }
    __syncthreads();

    const int wave  = tid >> 5;
    const int lane  = tid & 31;
    const int n     = lane & 15;                 // N column (0..15), valid < 8
    const int khalf = lane >> 4;                 // 0: K {0,1}, 1: K {2,3}
    const int mrow  = wave * 16 + n;             // A row held by this lane

    const float bn = (n < OUTN) ? bias[n] : 0.0f;
    v8f c;
    #pragma unroll
    for (int v = 0; v < 8; ++v) c[v] = bn;       // bias into accumulator

    const float* arow = avg + (size_t)mrow * DD + 2 * khalf;

    for (int k0 = 0; k0 < DD; k0 += 4) {
        v2f a = *(const v2f*)(arow + k0);        // A fragment (2 consecutive K)

        const int kbase = k0 + 2 * khalf;
        const float b0 = wt[(kbase + 0) * 8 + (n & 7)];
        const float b1 = wt[(kbase + 1) * 8 + (n & 7)];
        v2f bf;
        bf[0] = (n < OUTN) ? b0 : 0.0f;          // cndmask, no EXEC change
        bf[1] = (n < OUTN) ? b1 : 0.0f;

        // D = A x B + C   (v_wmma_f32_16x16x4_f32)
        c = __builtin_amdgcn_wmma_f32_16x16x4_f32(
                /*neg_a=*/false, a, /*neg_b=*/false, bf,
                /*c_mod=*/(short)0, c, /*reuse_a=*/false, /*reuse_b=*/false);
    }

    #pragma unroll
    for (int v = 0; v < 8; ++v) {
        const int row = wave * 16 + v + 8 * khalf;
        if (n < OUTN) out[row * OUTN + n] = c[v];
    }
}

// ---------------------------------------------------------------------------
extern "C" void kernel_launch(void* const* d_in, const int* in_sizes, int n_in,
                              void* d_out, int out_size, void* d_ws, size_t ws_size,
                              hipStream_t stream) {
    const float* x       = (const float*)d_in[0];   // [B,T,D] f32
    const int*   lengths = (const int*)  d_in[1];   // [B] int
    const float* W       = (const float*)d_in[2];   // [OUT,D] f32
    const float* bias    = (const float*)d_in[3];   // [OUT] f32
    float*       out     = (float*)d_out;           // [B,OUT] f32
    float*       avg     = (float*)d_ws;            // [B,D] f32 scratch (256 KB)

    dim3 grid1(DD / 128, BB);                       // 8 x 64 = 512 workgroups
    ragged_mean_kernel<<<grid1, 256, 0, stream>>>(x, lengths, avg);

    gemm_wmma_f32_kernel<<<1, 128, 0, stream>>>(avg, W, bias, out);
}